// AttentionModel_50809463112286
// MI455X (gfx1250) — compile-verified
//
#include <hip/hip_runtime.h>

// Problem constants (from reference)
#define NB 8
#define NT 2000
#define NH 256
#define NF 257
#define ALEN 128
#define SS 160   // LDS score-slab row stride (>=144, even for v2f loads)

typedef float v2f __attribute__((ext_vector_type(2)));
typedef float v8f __attribute__((ext_vector_type(8)));

// D(16x16 f32) = A(16x4 f32) * B(4x16 f32) + C   -- V_WMMA_F32_16X16X4_F32
__device__ __forceinline__ v8f wmma4(v2f a, v2f b, v8f c) {
  return __builtin_amdgcn_wmma_f32_16x16x4_f32(false, a, false, b, (short)0, c,
                                               false, false);
}

__global__ __launch_bounds__(128)
void attn_fused_kernel(const float* __restrict__ Kin, const float* __restrict__ Qin,
                       const float* __restrict__ Wsc, const float* __restrict__ Wen,
                       const float* __restrict__ ben, const float* __restrict__ Wmk,
                       const float* __restrict__ bmk, float* __restrict__ Out)
{
  __shared__ __align__(16) float qtile[16 * NH];  // q rows (concat operand)
  __shared__ __align__(16) float stile[16 * NH];  // qs, later reused for c
  __shared__ __align__(16) float otile[16 * NH];  // tanh output
  __shared__ __align__(16) float sbuf[16 * SS];   // banded raw scores -> weights
  __shared__ float wmax[4 * 16];
  __shared__ float rowmax[16];
  __shared__ float rowsum[16];
  __shared__ float rscale[16];

  const int ttile = blockIdx.x;           // 0..124 (T/16)
  const int bat   = blockIdx.y;           // 0..7
  const int t0    = ttile * 16;
  const int tid   = threadIdx.x;
  const int wave  = tid >> 5;
  const int lane  = tid & 31;
  const int half  = lane >> 4;            // wave32: lanes 0-15 / 16-31
  const int l16   = lane & 15;
  const int koff  = 2 * half;             // A/B .x holds K = kb + 2*half

  const float* qbase = Qin + ((size_t)bat * NT + t0) * NH;
  const float* kbase = Kin + (size_t)bat * NT * NH;

  // ---- Stage 1: q tile (contiguous 16x256 block) ----
  for (int i = tid; i < 16 * NH; i += 128) qtile[i] = qbase[i];
  __syncthreads();

  // ---- Stage 2a: qs = q @ W_score^T  -> stile ----
  for (int nt = wave; nt < NH / 16; nt += 4) {
    const int g0 = nt * 16;
    v8f acc = {};
    for (int kb = 0; kb < NH; kb += 4) {
      const int kk = kb + koff;
      v2f a = *(const v2f*)&qtile[l16 * NH + kk];
      v2f b = *(const v2f*)&Wsc[(size_t)(g0 + l16) * NH + kk];
      acc = wmma4(a, b, acc);
    }
#pragma unroll
    for (int r = 0; r < 8; ++r) stile[(r + 8 * half) * NH + g0 + l16] = acc[r];
  }
  __syncthreads();

  // ---- Stage 2b: full-row scores; track row max; stash band tiles ----
  const int st_lo = (ttile > 8) ? (ttile - 8) : 0;   // band window tile start
  float rmax[8];
#pragma unroll
  for (int r = 0; r < 8; ++r) rmax[r] = -3.4e38f;

  for (int st = wave; st < NT / 16; st += 4) {
    const int s0 = st * 16;
    v8f acc = {};
    for (int kb = 0; kb < NH; kb += 4) {
      const int kk = kb + koff;
      v2f a = *(const v2f*)&stile[l16 * NH + kk];                  // qs rows
      v2f b = *(const v2f*)&kbase[(size_t)(s0 + l16) * NH + kk];   // k^T cols
      acc = wmma4(a, b, acc);
    }
#pragma unroll
    for (int r = 0; r < 8; ++r) rmax[r] = fmaxf(rmax[r], acc[r]);
    if (st >= st_lo && st <= ttile) {
      const int c0 = (st - st_lo) * 16;
#pragma unroll
      for (int r = 0; r < 8; ++r) sbuf[(r + 8 * half) * SS + c0 + l16] = acc[r];
    }
  }

  // row max: reduce over 16 columns within each lane-half, then across waves
#pragma unroll
  for (int r = 0; r < 8; ++r) {
    float v = rmax[r];
    v = fmaxf(v, __shfl_xor(v, 1, 32));
    v = fmaxf(v, __shfl_xor(v, 2, 32));
    v = fmaxf(v, __shfl_xor(v, 4, 32));
    v = fmaxf(v, __shfl_xor(v, 8, 32));
    rmax[r] = v;
  }
  if (l16 == 0) {
#pragma unroll
    for (int r = 0; r < 8; ++r) wmax[wave * 16 + 8 * half + r] = rmax[r];
  }
  __syncthreads();
  if (tid < 16) {
    float m = wmax[tid];
    m = fmaxf(m, wmax[16 + tid]);
    m = fmaxf(m, wmax[32 + tid]);
    m = fmaxf(m, wmax[48 + tid]);
    rowmax[tid] = m;
    rowsum[tid] = 0.0f;
  }
  __syncthreads();

  // ---- Stage 3: banded exp + row sums ----
  const int ncols = (ttile - st_lo + 1) * 16;  // <= 144
  const int s0b   = st_lo * 16;
  {
    const int row = tid & 15;
    const int t   = t0 + row;
    float psum = 0.0f;
    for (int c = tid >> 4; c < ncols; c += 8) {
      const int s = s0b + c;
      float w = 0.0f;
      if (s <= t && s >= t - ALEN) w = __expf(sbuf[row * SS + c] - rowmax[row]);
      sbuf[row * SS + c] = w;
      psum += w;
    }
    atomicAdd(&rowsum[row], psum);
  }
  __syncthreads();
  if (tid < 16) rscale[tid] = 1.0f / (rowsum[tid] + 1e-30f);
  __syncthreads();

  // ---- Stage 4: c = (w @ k_window) * rscale  -> stile (qs dead, reuse) ----
  for (int nt = wave; nt < NH / 16; nt += 4) {
    const int h0 = nt * 16;
    v8f acc = {};
    for (int kb = 0; kb < ncols; kb += 4) {
      const int kk = kb + koff;
      v2f a = *(const v2f*)&sbuf[l16 * SS + kk];
      v2f b;
      b.x = kbase[(size_t)(s0b + kk)     * NH + h0 + l16];
      b.y = kbase[(size_t)(s0b + kk + 1) * NH + h0 + l16];
      acc = wmma4(a, b, acc);
    }
#pragma unroll
    for (int r = 0; r < 8; ++r)
      stile[(r + 8 * half) * NH + h0 + l16] = acc[r] * rscale[r + 8 * half];
  }
  __syncthreads();

  // ---- Stage 5: out = tanh([c,q] @ W_enh^T + b_enh) -> otile ----
  for (int nt = wave; nt < NH / 16; nt += 4) {
    const int j0 = nt * 16;
    v8f acc = {};
    for (int kb = 0; kb < NH; kb += 4) {          // c half of concat
      const int kk = kb + koff;
      v2f a = *(const v2f*)&stile[l16 * NH + kk];
      v2f b = *(const v2f*)&Wen[(size_t)(j0 + l16) * (2 * NH) + kk];
      acc = wmma4(a, b, acc);
    }
    for (int kb = 0; kb < NH; kb += 4) {          // q half of concat
      const int kk = kb + koff;
      v2f a = *(const v2f*)&qtile[l16 * NH + kk];
      v2f b = *(const v2f*)&Wen[(size_t)(j0 + l16) * (2 * NH) + NH + kk];
      acc = wmma4(a, b, acc);
    }
    const float bias = ben[j0 + l16];
#pragma unroll
    for (int r = 0; r < 8; ++r)
      otile[(r + 8 * half) * NH + j0 + l16] = tanhf(acc[r] + bias);
  }
  __syncthreads();

  // ---- Stage 6: mask = sigmoid(out @ W_mask^T + b_mask) -> global ----
  for (int ft = wave; ft < 17; ft += 4) {         // 257 -> 17 tiles, last padded
    const int f0 = ft * 16;
    const int f  = f0 + l16;
    v8f acc = {};
    for (int kb = 0; kb < NH; kb += 4) {
      const int kk = kb + koff;
      v2f a = *(const v2f*)&otile[l16 * NH + kk];
      v2f b = {0.0f, 0.0f};
      if (f < NF) b = *(const v2f*)&Wmk[(size_t)f * NH + kk];
      acc = wmma4(a, b, acc);
    }
    if (f < NF) {
      const float bias = bmk[f];
#pragma unroll
      for (int r = 0; r < 8; ++r) {
        const int t = t0 + r + 8 * half;
        const float v = acc[r] + bias;
        Out[((size_t)bat * NT + t) * NF + f] = 1.0f / (1.0f + __expf(-v));
      }
    }
  }
}

extern "C" void kernel_launch(void* const* d_in, const int* in_sizes, int n_in,
                              void* d_out, int out_size, void* d_ws, size_t ws_size,
                              hipStream_t stream) {
  (void)in_sizes; (void)n_in; (void)d_ws; (void)ws_size; (void)out_size;
  const float* k   = (const float*)d_in[0];
  const float* q   = (const float*)d_in[1];
  const float* Wsc = (const float*)d_in[2];
  const float* Wen = (const float*)d_in[3];
  const float* ben = (const float*)d_in[4];
  const float* Wmk = (const float*)d_in[5];
  const float* bmk = (const float*)d_in[6];
  dim3 grid(NT / 16, NB);
  attn_fused_kernel<<<grid, 128, 0, stream>>>(k, q, Wsc, Wen, ben, Wmk, bmk,
                                              (float*)d_out);
}